// LikelihoodRatioEstimator_64209761075845
// MI455X (gfx1250) — compile-verified
//
#include <hip/hip_runtime.h>
#include <hip/hip_bf16.h>
#include <math.h>

typedef float v2f __attribute__((ext_vector_type(2)));
typedef float v8f __attribute__((ext_vector_type(8)));
typedef int   v4i __attribute__((vector_size(16)));   // matches builtin param type

#define N_EMB 8192
#define D_EMB 512
#define PITCH 36           // LDS row pitch (floats): 144B rows -> 16B-aligned, conflict-free b64 frags
#define NKT   (D_EMB / 32) // 16 k-tiles of 32

#if defined(__has_builtin)
#  if __has_builtin(__builtin_amdgcn_global_load_async_to_lds_b128)
#    define USE_ASYNC_LDS 1
#  endif
#endif
#ifndef USE_ASYNC_LDS
#  define USE_ASYNC_LDS 0
#endif

#if USE_ASYNC_LDS
#  if __has_builtin(__builtin_amdgcn_s_wait_asynccnt)
#    define WAIT_ASYNC(n) __builtin_amdgcn_s_wait_asynccnt(n)
#  else
#    define WAIT_ASYNC(n) asm volatile("s_wait_asynccnt %0" ::"i"(n) : "memory")
#  endif
__device__ __forceinline__ void async_cp16(const float* g, float* l) {
    // generic pointers; compiler infers global/LDS address spaces
    __builtin_amdgcn_global_load_async_to_lds_b128((v4i*)g, (v4i*)l, 0, 0);
}
#endif

// ---------------------------------------------------------------------------
// Row squared-norms: one wave per row (wave32), shuffle reduce.
// ---------------------------------------------------------------------------
__global__ __launch_bounds__(256) void rownorm_kernel(const float* __restrict__ ctx,
                                                      const float* __restrict__ tgt,
                                                      float* __restrict__ c2,
                                                      float* __restrict__ t2) {
    int gw   = blockIdx.x * 8 + (threadIdx.x >> 5);
    int lane = threadIdx.x & 31;
    const float* src = (gw < N_EMB) ? ctx : tgt;
    float*       dst = (gw < N_EMB) ? c2  : t2;
    int row = gw & (N_EMB - 1);
    const float* p = src + (size_t)row * D_EMB;
    float s = 0.f;
#pragma unroll
    for (int j = 0; j < D_EMB / 32; ++j) {
        float v = p[lane + j * 32];
        s = fmaf(v, v, s);
    }
#pragma unroll
    for (int off = 16; off; off >>= 1) s += __shfl_xor(s, off, 32);
    if (lane == 0) dst[row] = s;
}

__global__ void init_kernel(double* accum) {
    if (threadIdx.x < 4) accum[threadIdx.x] = 0.0;
}

// ---------------------------------------------------------------------------
// Fused Gram + Cauchy-logit reduction. 128x128 tile/block, 8 waves x (32x64),
// v_wmma_f32_16x16x4_f32, double-buffered LDS with async-to-LDS (or pipelined
// register staging fallback).
// PASS 0: sum 1/(1+d2), sum log1p(d2) off-diag; store diagonal d2.
// PASS 1: sum 1/(1+K(1+d2)), count d2 < T off-diag.
// ---------------------------------------------------------------------------
template <int PASS>
__global__ __launch_bounds__(256)
void cauchy_gemm_kernel(const float* __restrict__ ctx, const float* __restrict__ tgt,
                        const float* __restrict__ c2,  const float* __restrict__ t2,
                        float* __restrict__ pos_d2, double* __restrict__ accum,
                        const float* __restrict__ fscal) {
    __shared__ float As[2][128 * PITCH];
    __shared__ float Bs[2][128 * PITCH];
    __shared__ float red0[8];
    __shared__ float red1[8];

    const int tid  = threadIdx.x;
    const int lane = tid & 31;
    const int wave = tid >> 5;
    const int hi   = lane >> 4;
    const int lm   = lane & 15;
    const int wRow = wave & 3;
    const int wCol = wave >> 2;

    const int rowBase = blockIdx.y * 128;
    const int colBase = blockIdx.x * 128;
    const float* gA = ctx + (size_t)rowBase * D_EMB;
    const float* gB = tgt + (size_t)colBase * D_EMB;

    // staging coordinates: each thread owns a fixed 16B column, 4 row-strips
    const int r0 = tid >> 3;
    const int c4 = (tid & 7) << 2;

    v8f acc[2][4];
#pragma unroll
    for (int rt = 0; rt < 2; ++rt)
#pragma unroll
        for (int ct = 0; ct < 4; ++ct) acc[rt][ct] = (v8f){0.f,0.f,0.f,0.f,0.f,0.f,0.f,0.f};

    auto compute_tile = [&](const float* Ab, const float* Bb) {
#pragma unroll
        for (int kk = 0; kk < 32; kk += 4) {
            v2f af[2], bf[4];
#pragma unroll
            for (int rt = 0; rt < 2; ++rt)
                af[rt] = *(const v2f*)(Ab + (wRow * 32 + rt * 16 + lm) * PITCH + kk + 2 * hi);
#pragma unroll
            for (int ct = 0; ct < 4; ++ct)
                bf[ct] = *(const v2f*)(Bb + (wCol * 64 + ct * 16 + lm) * PITCH + kk + 2 * hi);
#pragma unroll
            for (int rt = 0; rt < 2; ++rt)
#pragma unroll
                for (int ct = 0; ct < 4; ++ct)
                    acc[rt][ct] = __builtin_amdgcn_wmma_f32_16x16x4_f32(
                        false, af[rt], false, bf[ct], (short)0, acc[rt][ct], false, false);
        }
    };

#if USE_ASYNC_LDS
    auto issue_tile = [&](int kt, int buf) {
#pragma unroll
        for (int it = 0; it < 4; ++it) {
            int r = r0 + it * 32;
            async_cp16(gA + (size_t)r * D_EMB + kt + c4, &As[buf][r * PITCH + c4]);
            async_cp16(gB + (size_t)r * D_EMB + kt + c4, &Bs[buf][r * PITCH + c4]);
        }
    };
    issue_tile(0, 0);
    for (int t = 0; t < NKT; ++t) {
        const int buf = t & 1;
        if (t + 1 < NKT) {
            issue_tile((t + 1) * 32, buf ^ 1);   // overlap next copy with this stage
            WAIT_ASYNC(8);                       // oldest 8 (this tile) complete
        } else {
            WAIT_ASYNC(0);
        }
        __syncthreads();
        compute_tile(&As[buf][0], &Bs[buf][0]);
        __syncthreads();                         // all readers done before buf refilled
    }
#else
    float4 ra[4], rb[4];
    auto load_regs = [&](int kt) {
#pragma unroll
        for (int it = 0; it < 4; ++it) {
            int r = r0 + it * 32;
            ra[it] = *(const float4*)(gA + (size_t)r * D_EMB + kt + c4);
            rb[it] = *(const float4*)(gB + (size_t)r * D_EMB + kt + c4);
        }
    };
    load_regs(0);
    for (int t = 0; t < NKT; ++t) {
        const int buf = t & 1;
#pragma unroll
        for (int it = 0; it < 4; ++it) {
            int r = r0 + it * 32;
            *(float4*)(&As[buf][r * PITCH + c4]) = ra[it];
            *(float4*)(&Bs[buf][r * PITCH + c4]) = rb[it];
        }
        __syncthreads();
        if (t + 1 < NKT) load_regs((t + 1) * 32);  // global latency hidden behind WMMAs
        compute_tile(&As[buf][0], &Bs[buf][0]);
        // single barrier/iter: next store targets the other buffer
    }
#endif

    // Epilogue: d2 = max(c2 + t2 - 2*dot, 0)
    float Kc = 0.f, Tc = 0.f;
    if (PASS == 1) { Kc = fscal[0]; Tc = fscal[1]; }
    float s0 = 0.f, s1 = 0.f;
#pragma unroll
    for (int rt = 0; rt < 2; ++rt) {
#pragma unroll
        for (int ct = 0; ct < 4; ++ct) {
            int n = colBase + wCol * 64 + ct * 16 + lm;
            float t2n = t2[n];
#pragma unroll
            for (int v = 0; v < 8; ++v) {
                int m = rowBase + wRow * 32 + rt * 16 + hi * 8 + v;
                float dot = acc[rt][ct][v];
                float d2 = fmaxf(c2[m] + t2n - 2.f * dot, 0.f);
                if (PASS == 0) {
                    if (m == n) {
                        pos_d2[m] = d2;
                    } else {
                        s0 += 1.f / (1.f + d2);   // exp(logit)
                        s1 += log1pf(d2);         // -logit
                    }
                } else {
                    if (m != n) {
                        s0 += 1.f / (1.f + Kc * (1.f + d2));   // sigmoid(logit - lb)
                        s1 += (d2 < Tc) ? 1.f : 0.f;            // logit > lb
                    }
                }
            }
        }
    }
#pragma unroll
    for (int off = 16; off; off >>= 1) {
        s0 += __shfl_xor(s0, off, 32);
        s1 += __shfl_xor(s1, off, 32);
    }
    if (lane == 0) { red0[wave] = s0; red1[wave] = s1; }
    __syncthreads();
    if (tid == 0) {
        float a0 = 0.f, a1 = 0.f;
#pragma unroll
        for (int w = 0; w < 8; ++w) { a0 += red0[w]; a1 += red1[w]; }
        if (PASS == 0) {
            atomicAdd(&accum[0], (double)a0);
            atomicAdd(&accum[1], (double)a1);
        } else {
            atomicAdd(&accum[2], (double)a0);
            atomicAdd(&accum[3], (double)a1);
        }
    }
}

// ---------------------------------------------------------------------------
__global__ void lb_kernel(const double* __restrict__ accum, double* __restrict__ dscal,
                          float* __restrict__ fscal) {
    double sum_exp = accum[0];
    double cntneg  = (double)N_EMB * (double)(N_EMB - 1);
    double lb = log(sum_exp) - log(cntneg);
    double K  = sum_exp / cntneg;          // e^lb
    double T  = cntneg / sum_exp - 1.0;    // e^-lb - 1
    dscal[0] = lb; dscal[1] = K; dscal[2] = T;
    fscal[0] = (float)K; fscal[1] = (float)T;
}

// ---------------------------------------------------------------------------
__global__ __launch_bounds__(256)
void final_kernel(const float* __restrict__ pos_d2, const double* __restrict__ accum,
                  const double* __restrict__ dscal, float* __restrict__ out) {
    __shared__ float r0[8], r1[8], r2[8];
    const int tid = threadIdx.x, lane = tid & 31, wave = tid >> 5;
    float Kc = (float)dscal[1], Tc = (float)dscal[2];
    float sl = 0.f, ss = 0.f, sc = 0.f;
    for (int i = tid; i < N_EMB; i += 256) {
        float d2 = pos_d2[i];
        sl += -log1pf(d2);
        ss += 1.f / (1.f + Kc * (1.f + d2));
        sc += (d2 < Tc) ? 1.f : 0.f;
    }
#pragma unroll
    for (int off = 16; off; off >>= 1) {
        sl += __shfl_xor(sl, off, 32);
        ss += __shfl_xor(ss, off, 32);
        sc += __shfl_xor(sc, off, 32);
    }
    if (lane == 0) { r0[wave] = sl; r1[wave] = ss; r2[wave] = sc; }
    __syncthreads();
    if (tid == 0) {
        double tsl = 0, tss = 0, tsc = 0;
#pragma unroll
        for (int w = 0; w < 8; ++w) { tsl += r0[w]; tss += r1[w]; tsc += r2[w]; }
        const double Nf = (double)N_EMB;
        const double CN = (double)N_EMB * (double)(N_EMB - 1);
        double lb = dscal[0], K = dscal[1];
        double sum_exp = accum[0], sum_log1p = accum[1];
        double sum_sig_neg = accum[2], cnt_neg_gt = accum[3];

        double pos_mean  = tsl / Nf - lb;
        double repulsion = sum_exp / (K * CN);
        double loss      = -pos_mean + repulsion;
        double neg_mean  = (-sum_log1p) / CN - lb;
        double sig_pos   = tss / Nf;
        double sig_neg   = sum_sig_neg / CN;
        double TP = tsc / Nf,        FN = 1.0 - TP;
        double FP = cnt_neg_gt / CN, TN = 1.0 - FP;
        double accuracy  = 0.5 * (TP + TN);
        double precision = TP / (TP + FP);
        double npv       = TN / (TN + FN);
        double apv       = 0.5 * (precision + npv);

        out[0]  = (float)loss;
        out[1]  = (float)pos_mean;
        out[2]  = (float)neg_mean;
        out[3]  = (float)sig_pos;
        out[4]  = (float)sig_neg;
        out[5]  = (float)lb;
        out[6]  = (float)accuracy;
        out[7]  = (float)precision;
        out[8]  = (float)npv;
        out[9]  = (float)apv;
        out[10] = (float)TP;
        out[11] = (float)TN;
    }
}

extern "C" void kernel_launch(void* const* d_in, const int* in_sizes, int n_in,
                              void* d_out, int out_size, void* d_ws, size_t ws_size,
                              hipStream_t stream) {
    const float* ctx = (const float*)d_in[0];
    const float* tgt = (const float*)d_in[1];
    float* out = (float*)d_out;

    char* ws = (char*)d_ws;
    float*  c2     = (float*)(ws);
    float*  t2     = (float*)(ws + 32768);
    float*  pos_d2 = (float*)(ws + 65536);
    double* accum  = (double*)(ws + 98304);
    double* dscal  = accum + 4;
    float*  fscal  = (float*)(dscal + 3);

    init_kernel<<<1, 32, 0, stream>>>(accum);
    rownorm_kernel<<<2 * N_EMB / 8, 256, 0, stream>>>(ctx, tgt, c2, t2);

    dim3 grid(N_EMB / 128, N_EMB / 128);
    cauchy_gemm_kernel<0><<<grid, 256, 0, stream>>>(ctx, tgt, c2, t2, pos_d2, accum, fscal);
    lb_kernel<<<1, 1, 0, stream>>>(accum, dscal, fscal);
    cauchy_gemm_kernel<1><<<grid, 256, 0, stream>>>(ctx, tgt, c2, t2, pos_d2, accum, fscal);
    final_kernel<<<1, 256, 0, stream>>>(pos_d2, accum, dscal, out);
}